// SimpleGCN_19997367730675
// MI455X (gfx1250) — compile-verified
//
#include <hip/hip_runtime.h>

#define DIM 128

typedef float v2f __attribute__((ext_vector_type(2)));
typedef float v8f __attribute__((ext_vector_type(8)));

// ---------------------------------------------------------------------------
// Zero the int counter arrays (out_cnt, in_cnt, cursor). Runs every launch so
// graph replay is self-contained.
// ---------------------------------------------------------------------------
__global__ void zero_int_kernel(int* __restrict__ a, int* __restrict__ b,
                                int* __restrict__ c, int n) {
    int i = blockIdx.x * blockDim.x + threadIdx.x;
    if (i < n) { a[i] = 0; b[i] = 0; c[i] = 0; }
}

// Degree counting: one int atomic per endpoint (1.6M each, not 210M).
__global__ void count_edges_kernel(const int* __restrict__ src, const int* __restrict__ dst,
                                   int* __restrict__ out_cnt, int* __restrict__ in_cnt, int e) {
    int i = blockIdx.x * blockDim.x + threadIdx.x;
    if (i < e) {
        atomicAdd(&out_cnt[src[i]], 1);
        atomicAdd(&in_cnt[dst[i]], 1);
    }
}

// norm = rsqrt(deg + 1)   (+1 = self loop)
__global__ void norm_kernel(const int* __restrict__ out_cnt, const int* __restrict__ in_cnt,
                            float* __restrict__ norm_src, float* __restrict__ norm_dst, int n) {
    int i = blockIdx.x * blockDim.x + threadIdx.x;
    if (i < n) {
        norm_src[i] = rsqrtf((float)(out_cnt[i] + 1));
        norm_dst[i] = rsqrtf((float)(in_cnt[i] + 1));
    }
}

// ---------------------------------------------------------------------------
// Exclusive scan of in-degree counts -> CSR row offsets. Single 1024-thread
// block: each thread serially sums a chunk, LDS Hillis-Steele over partials,
// then writes its chunk's running offsets. 50K elements: microseconds.
// ---------------------------------------------------------------------------
__global__ void scan_offsets_kernel(const int* __restrict__ cnt, int* __restrict__ offs, int n) {
    __shared__ int partial[1024];
    int tid = threadIdx.x;
    int chunk = (n + 1023) / 1024;
    int start = tid * chunk;
    int end   = start + chunk; if (end > n) end = n;

    int sum = 0;
    for (int i = start; i < end; ++i) sum += cnt[i];
    partial[tid] = sum;
    __syncthreads();

    for (int off = 1; off < 1024; off <<= 1) {
        int v = 0;
        if (tid >= off) v = partial[tid - off];
        __syncthreads();
        partial[tid] += v;
        __syncthreads();
    }

    int running = (tid == 0) ? 0 : partial[tid - 1];   // exclusive base
    for (int i = start; i < end; ++i) {
        offs[i] = running;
        running += cnt[i];
    }
    if (tid == 1023) offs[n] = partial[1023];          // total edge count
}

// CSR fill: one cursor atomic per edge.
__global__ void fill_csr_kernel(const int* __restrict__ src, const int* __restrict__ dst,
                                const int* __restrict__ offs, int* __restrict__ cursor,
                                int* __restrict__ edge_src, int e) {
    int i = blockIdx.x * blockDim.x + threadIdx.x;
    if (i < e) {
        int d = dst[i];
        int p = atomicAdd(&cursor[d], 1);
        edge_src[offs[d] + p] = src[i];
    }
}

// ---------------------------------------------------------------------------
// hs = h * norm_src (row scale). One thread per float4. Safe when h == hs.
// ---------------------------------------------------------------------------
__global__ void scale_kernel(const float* __restrict__ h, const float* __restrict__ norm_src,
                             float* __restrict__ hs, int n) {
    int t = blockIdx.x * blockDim.x + threadIdx.x;
    if (t >= n * 32) return;
    int node = t >> 5;
    float s = norm_src[node];
    float4 v = ((const float4*)h)[t];
    v.x *= s; v.y *= s; v.z *= s; v.w *= s;
    ((float4*)hs)[t] = v;
}

// ---------------------------------------------------------------------------
// Pull aggregation: one wave per destination node. Accumulator starts at the
// self-loop row; each incoming edge is a coalesced float4 gather of hs[src]
// (L2-resident: hs = 25.6 MB << 192 MB L2). norm_dst fused into the store.
// ZERO atomics in this hot loop.
// ---------------------------------------------------------------------------
__global__ void gather_agg_kernel(const float* __restrict__ hs, const int* __restrict__ offs,
                                  const int* __restrict__ edge_src,
                                  const float* __restrict__ norm_dst,
                                  float* __restrict__ agg, int n) {
    int wave = threadIdx.x >> 5;
    int node = blockIdx.x * (blockDim.x >> 5) + wave;
    if (node >= n) return;
    int lane = threadIdx.x & 31;

    float4 acc = ((const float4*)(hs + (size_t)node * DIM))[lane];   // self loop
    int beg = offs[node];
    int end = offs[node + 1];
    #pragma unroll 4
    for (int j = beg; j < end; ++j) {
        int s = edge_src[j];                                          // wave-uniform
        float4 v = ((const float4*)(hs + (size_t)s * DIM))[lane];
        acc.x += v.x; acc.y += v.y; acc.z += v.z; acc.w += v.w;
    }
    float nd = norm_dst[node];
    acc.x *= nd; acc.y *= nd; acc.z *= nd; acc.w *= nd;
    ((float4*)(agg + (size_t)node * DIM))[lane] = acc;
}

// ---------------------------------------------------------------------------
// out = relu?( agg @ W + bias )  via V_WMMA_F32_16X16X4_F32 (exact fp32).
// One wave owns a 16(M) x 128(N) output stripe (8 accumulators).
//   A (16x4):  lane = 16*half + sub -> M = sub, K = k0 + 2*half + j
//   B (4x16):  same lanes           -> N = sub, K = k0 + 2*half + j
//   C/D:       VGPR i holds row (i + 8*half), col sub
// EXEC stays all-ones through the WMMA loop (guards only at the store).
// ---------------------------------------------------------------------------
__global__ void gemm_bias_kernel(const float* __restrict__ agg,
                                 const float* __restrict__ W, const float* __restrict__ bias,
                                 float* __restrict__ out, int n, int nTiles, int relu) {
    int wave = threadIdx.x >> 5;
    int tile = blockIdx.x * (blockDim.x >> 5) + wave;
    if (tile >= nTiles) return;

    int lane = threadIdx.x & 31;
    int half = lane >> 4;
    int sub  = lane & 15;

    int rowA = tile * 16 + sub;
    if (rowA > n - 1) rowA = n - 1;            // clamp, no EXEC divergence
    const float* Arow = agg + (size_t)rowA * DIM;

    v8f c[8] = {};

    for (int k0 = 0; k0 < DIM; k0 += 4) {
        int ka = k0 + 2 * half;
        v2f a;
        a[0] = Arow[ka + 0];
        a[1] = Arow[ka + 1];
        #pragma unroll
        for (int nt = 0; nt < 8; ++nt) {
            v2f b;
            b[0] = W[(size_t)(ka + 0) * DIM + nt * 16 + sub];
            b[1] = W[(size_t)(ka + 1) * DIM + nt * 16 + sub];
            c[nt] = __builtin_amdgcn_wmma_f32_16x16x4_f32(
                false, a, false, b, (short)0, c[nt], false, false);
        }
    }

    #pragma unroll
    for (int nt = 0; nt < 8; ++nt) {
        int col = nt * 16 + sub;
        float bv = bias[col];
        #pragma unroll
        for (int i = 0; i < 8; ++i) {
            int row = tile * 16 + i + 8 * half;
            float val = c[nt][i] + bv;
            if (relu) val = fmaxf(val, 0.0f);
            if (row < n) out[(size_t)row * DIM + col] = val;
        }
    }
}

// ---------------------------------------------------------------------------
// Launch: degrees -> norms -> CSR (built once, shared by both layers)
//         -> [scale, pull-gather, wmma-gemm] x 2 layers
// ---------------------------------------------------------------------------
extern "C" void kernel_launch(void* const* d_in, const int* in_sizes, int n_in,
                              void* d_out, int out_size, void* d_ws, size_t ws_size,
                              hipStream_t stream) {
    const float* feat = (const float*)d_in[0];
    const float* W1   = (const float*)d_in[1];
    const float* b1   = (const float*)d_in[2];
    const float* W2   = (const float*)d_in[3];
    const float* b2   = (const float*)d_in[4];
    const int*   src  = (const int*)d_in[5];
    const int*   dst  = (const int*)d_in[6];

    int n = in_sizes[0] / DIM;   // 50000
    int e = in_sizes[5];         // 1600000

    // Workspace carve-up (4-byte elements, each region rounded to 16 B).
    char* base = (char*)d_ws;
    size_t o = 0;
    auto alloc = [&](size_t elems) { char* p = base + o * 4; o += (elems + 3) & ~(size_t)3; return p; };
    float* norm_src = (float*)alloc(n);
    float* norm_dst = (float*)alloc(n);
    int*   out_cnt  = (int*)  alloc(n);
    int*   in_cnt   = (int*)  alloc(n);
    int*   cursor   = (int*)  alloc(n);
    int*   offs     = (int*)  alloc(n + 1);
    int*   edge_src = (int*)  alloc(e);
    float* hs       = (float*)alloc((size_t)n * DIM);   // reused as layer-1 output
    float* agg      = (float*)alloc((size_t)n * DIM);
    float* out      = (float*)d_out;

    dim3 blk(256);
    int nBlkN  = (n + 255) / 256;
    int nBlkE  = (e + 255) / 256;
    int nTiles = (n + 15) / 16;
    int gemmGrid   = (nTiles + 7) / 8;        // 8 waves / block
    int scaleGrid  = (n * 32 + 255) / 256;
    int gatherGrid = (n + 7) / 8;             // wave per node, 8 waves / block

    // degrees + norms + CSR (once)
    zero_int_kernel<<<nBlkN, blk, 0, stream>>>(out_cnt, in_cnt, cursor, n);
    count_edges_kernel<<<nBlkE, blk, 0, stream>>>(src, dst, out_cnt, in_cnt, e);
    norm_kernel<<<nBlkN, blk, 0, stream>>>(out_cnt, in_cnt, norm_src, norm_dst, n);
    scan_offsets_kernel<<<1, 1024, 0, stream>>>(in_cnt, offs, n);
    fill_csr_kernel<<<nBlkE, blk, 0, stream>>>(src, dst, offs, cursor, edge_src, e);

    // layer 1
    scale_kernel<<<scaleGrid, blk, 0, stream>>>(feat, norm_src, hs, n);
    gather_agg_kernel<<<gatherGrid, blk, 0, stream>>>(hs, offs, edge_src, norm_dst, agg, n);
    gemm_bias_kernel<<<gemmGrid, blk, 0, stream>>>(agg, W1, b1, hs, n, nTiles, 1);

    // layer 2 (scale is element-wise, in place safe: reads hs, writes hs)
    scale_kernel<<<scaleGrid, blk, 0, stream>>>(hs, norm_src, hs, n);
    gather_agg_kernel<<<gatherGrid, blk, 0, stream>>>(hs, offs, edge_src, norm_dst, agg, n);
    gemm_bias_kernel<<<gemmGrid, blk, 0, stream>>>(agg, W2, b2, out, n, nTiles, 0);
}